// LearnableDWT_24369644437793
// MI455X (gfx1250) — compile-verified
//
#include <hip/hip_runtime.h>
#include <hip/hip_bf16.h>

// LearnableDWT on MI455X (gfx1250):
//   x:[8,96,512,512] f32 -> 4 subbands [8,96,256,256] f32 (concatenated in d_out).
// Memory bound (0.9 flop/B vs 23.3 TB/s HBM): pure streaming kernel.
// Compute path: V_WMMA_F32_16X16X4_F32 — A (16 patches x 4 taps) is formed
// directly by two coalesced 128B row-segment loads (lanes 0-15 = top row
// pairs, lanes 16-31 = bottom row pairs), B holds the 4 learnable 2x2 filters
// in columns N=0..3. One WMMA replaces 16 scalar FMAs per patch.
// Inner loop is double-buffered: two A tiles load back-to-back so one load is
// always in flight while the previous tile's WMMA+stores retire, and the two
// tiles' masked stores share a single EXEC save/restore.

typedef __attribute__((ext_vector_type(2))) float v2f;
typedef __attribute__((ext_vector_type(4))) float v4f;
typedef __attribute__((ext_vector_type(8))) float v8f;

#define DWT_H   512
#define DWT_W   512
#define DWT_H2  256
#define DWT_W2  256

__global__ __launch_bounds__(256) void dwt_wmma_kernel(
    const float* __restrict__ x,
    const float* __restrict__ wll, const float* __restrict__ wlh,
    const float* __restrict__ whl, const float* __restrict__ whh,
    float* __restrict__ out, int C)
{
    const int lane = threadIdx.x & 31;
    const int wave = threadIdx.x >> 5;
    const int bc   = blockIdx.x;              // fused (b, c) index
    const int c    = bc % C;
    const int i    = blockIdx.y * 8 + wave;   // output row 0..255

    const int sub = lane & 15;                // candidate subband / patch id
    const int hi  = lane >> 4;                // 0 = lanes 0-15, 1 = lanes 16-31

    // ---- Build B matrix (4x16 f32, only N=0..3 used) ----
    // Layout: VGPR0 = row K0 (lanes 0-15) | row K1 (lanes 16-31)
    //         VGPR1 = row K2 (lanes 0-15) | row K3 (lanes 16-31)
    // Weight flat layout per channel: w[c*4 + p*2 + q], K order = (p,q) row-major,
    // matching A's K order (x00, x01, x10, x11).
    const float* wsel = (sub == 0) ? wll : (sub == 1) ? wlh
                      : (sub == 2) ? whl : whh;
    v2f bm;
    if (sub < 4) {
        bm.x = wsel[c * 4 + 0 + hi];   // K = 0 (hi=0) or K = 1 (hi=1)
        bm.y = wsel[c * 4 + 2 + hi];   // K = 2 (hi=0) or K = 3 (hi=1)
    } else {
        bm.x = 0.0f; bm.y = 0.0f;
    }

    // Input row for this half-wave: even row for lanes 0-15, odd row for 16-31.
    const float* rowbase =
        x + ((size_t)bc * DWT_H + (size_t)(2 * i + hi)) * DWT_W;

    const size_t Nsub   = (size_t)gridDim.x * DWT_H2 * DWT_W2;  // elems / subband
    const size_t outrow = ((size_t)bc * DWT_H2 + (size_t)i) * DWT_W2;

    // Store base for this lane (valid when sub < 4):
    // lane s       -> subband s, patches M=0..7  (cols j0+0..7)
    // lane 16+s    -> subband s, patches M=8..15 (cols j0+8..15)
    float* op = out + (size_t)sub * Nsub + outrow + (size_t)(hi * 8);

    #pragma unroll 2
    for (int j0 = 0; j0 < DWT_W2; j0 += 32) {
        // Issue both tile loads before consuming either: first WMMA waits on
        // loadcnt<=1 and the second load stays in flight behind it.
        v2f a0 = __builtin_nontemporal_load(
            (const v2f*)(rowbase + 2 * j0 + 2 * sub));
        v2f a1 = __builtin_nontemporal_load(
            (const v2f*)(rowbase + 2 * j0 + 32 + 2 * sub));

        v8f cz0 = {};
        v8f d0 = __builtin_amdgcn_wmma_f32_16x16x4_f32(
            false, a0, false, bm, (short)0, cz0, false, false);
        v8f cz1 = {};
        v8f d1 = __builtin_amdgcn_wmma_f32_16x16x4_f32(
            false, a1, false, bm, (short)0, cz1, false, false);

        if (sub < 4) {
            // D layout: lanes 0-15  VGPR r = D[M=r  ][N=lane]
            //           lanes 16-31 VGPR r = D[M=8+r][N=lane-16]
            v4f t0 = { d0[0], d0[1], d0[2], d0[3] };
            v4f t1 = { d0[4], d0[5], d0[6], d0[7] };
            v4f t2 = { d1[0], d1[1], d1[2], d1[3] };
            v4f t3 = { d1[4], d1[5], d1[6], d1[7] };
            __builtin_nontemporal_store(t0, (v4f*)(op + j0));
            __builtin_nontemporal_store(t1, (v4f*)(op + j0 + 4));
            __builtin_nontemporal_store(t2, (v4f*)(op + j0 + 16));
            __builtin_nontemporal_store(t3, (v4f*)(op + j0 + 20));
        }
    }
}

extern "C" void kernel_launch(void* const* d_in, const int* in_sizes, int n_in,
                              void* d_out, int out_size, void* d_ws, size_t ws_size,
                              hipStream_t stream) {
    const float* x   = (const float*)d_in[0];
    const float* wll = (const float*)d_in[1];
    const float* wlh = (const float*)d_in[2];
    const float* whl = (const float*)d_in[3];
    const float* whh = (const float*)d_in[4];
    float* out = (float*)d_out;

    const int C  = in_sizes[1] / 4;                       // wll is [C,2,2]
    const int BC = (int)((long long)in_sizes[0] / ((long long)DWT_H * DWT_W));

    dim3 grid(BC, DWT_H2 / 8, 1);   // 768 x 32 blocks
    dim3 block(256, 1, 1);          // 8 waves; 1 wave per output row
    dwt_wmma_kernel<<<grid, block, 0, stream>>>(x, wll, wlh, whl, whh, out, C);
}